// FMFMNeuron_42262478192713
// MI455X (gfx1250) — compile-verified
//
#include <hip/hip_runtime.h>

// LIF spiking-neuron scan, MI455X (gfx1250).
// Memory-bound: 402 MB traffic -> ~17us floor @ 23.3 TB/s. WMMA inapplicable
// (K=2,N=1 fused into a sequential scan). Use CDNA5 async-to-LDS streaming
// pipeline (global_load_async_to_lds_b64 + s_wait_asynccnt) for an 8-deep
// explicit prefetch pipeline; fallback is SW-pipelined regs + global_prefetch.

#ifndef __has_builtin
#define __has_builtin(x) 0
#endif

#if __has_builtin(__builtin_amdgcn_global_load_async_to_lds_b64) && \
    __has_builtin(__builtin_amdgcn_s_wait_asynccnt)
#define LIF_ASYNC 1
#else
#define LIF_ASYNC 0
#endif

#define LIF_BLOCK 256
#define LIF_STAGES 8  // 8-deep async pipeline: 8 slots * 256 thr * 8B = 16 KB LDS

// -------------------- async-to-LDS pipelined variant --------------------
#if LIF_ASYNC
// Builtin signature (from compiler diagnostic): pointers are to int2 vectors,
// global src in addrspace(1), LDS dst in addrspace(3), then imm offset, imm cpol.
typedef int v2i __attribute__((ext_vector_type(2)));
typedef __attribute__((address_space(1))) v2i* gptr_b64;
typedef __attribute__((address_space(3))) v2i* lptr_b64;

__global__ __launch_bounds__(LIF_BLOCK) void lif_async_kernel(
    const float* __restrict__ spikes, const float* __restrict__ w,
    float* __restrict__ out, int T, int B) {
  __shared__ float2 buf[LIF_STAGES][LIF_BLOCK];
  const int tid = threadIdx.x;
  const int b = blockIdx.x * LIF_BLOCK + tid;
  const float w0 = w[0];
  const float w1 = w[1];
  const float2* __restrict__ src = (const float2*)spikes + b;  // stride B per t
  float* __restrict__ dst = out + b;                           // stride B per t

  // Prime the pipeline with STAGES-1 async loads (per-lane 8B -> own LDS slot).
#pragma unroll
  for (int s = 0; s < LIF_STAGES - 1; ++s) {
    __builtin_amdgcn_global_load_async_to_lds_b64(
        (gptr_b64)(src + (size_t)s * B), (lptr_b64)&buf[s][tid], 0, 0);
  }

  float mem = 0.0f;
  float reset = 0.0f;  // reset_t == spk_{t-1}
  for (int t = 0; t < T; ++t) {
    // Keep exactly STAGES loads in flight; clamp the tail (dummy loads into
    // slots that are never consumed again) so the wait immediate is constant.
    const int tp = t + (LIF_STAGES - 1);
    const int tc = (tp < T) ? tp : (T - 1);
    __builtin_amdgcn_global_load_async_to_lds_b64(
        (gptr_b64)(src + (size_t)tc * B),
        (lptr_b64)&buf[tp & (LIF_STAGES - 1)][tid], 0, 0);
    // Async loads retire in order: <=7 outstanding => load for step t landed.
    __builtin_amdgcn_s_wait_asynccnt(LIF_STAGES - 1);

    const float2 sp = buf[t & (LIF_STAGES - 1)][tid];  // ds_load_b64
    const float c = fmaf(sp.x, w0, sp.y * w1);
    mem = fmaf(0.95f, mem, c) - reset;                 // reset * THRESH, THRESH=1
    const float spk = (mem > 1.0f) ? 1.0f : 0.0f;      // (mem_new - 1 > 0)
    dst[(size_t)t * B] = spk;
    reset = spk;
  }
}
#endif  // LIF_ASYNC

// -------------------- fallback: SW-pipelined registers --------------------
__global__ __launch_bounds__(LIF_BLOCK) void lif_plain_kernel(
    const float* __restrict__ spikes, const float* __restrict__ w,
    float* __restrict__ out, int T, int B) {
  const int b = blockIdx.x * LIF_BLOCK + threadIdx.x;
  const float w0 = w[0];
  const float w1 = w[1];
  const float2* __restrict__ src = (const float2*)spikes + b;
  float* __restrict__ dst = out + b;

  float mem = 0.0f;
  float reset = 0.0f;
  float2 cur = src[0];
  for (int t = 0; t < T; ++t) {
    if (t + 16 < T)  // warm L2 ~16 steps ahead (global_prefetch_b8)
      __builtin_prefetch((const void*)(src + (size_t)(t + 16) * B), 0, 0);
    float2 nxt = cur;
    if (t + 1 < T) nxt = src[(size_t)(t + 1) * B];  // load-use distance = 1 iter

    const float c = fmaf(cur.x, w0, cur.y * w1);
    mem = fmaf(0.95f, mem, c) - reset;
    const float spk = (mem > 1.0f) ? 1.0f : 0.0f;
    dst[(size_t)t * B] = spk;
    reset = spk;
    cur = nxt;
  }
}

extern "C" void kernel_launch(void* const* d_in, const int* in_sizes, int n_in,
                              void* d_out, int out_size, void* d_ws, size_t ws_size,
                              hipStream_t stream) {
  const float* spikes = (const float*)d_in[0];  // [T, B, 2] f32
  const float* w      = (const float*)d_in[1];  // [1, 2]   f32
  float* out          = (float*)d_out;          // [T, B, 1] f32

  const int B = 65536;               // reference batch (layout stride)
  const int T = out_size / B;        // 512

  dim3 block(LIF_BLOCK);
  dim3 grid(B / LIF_BLOCK);          // 256 workgroups x 8 waves = 2048 waves
#if LIF_ASYNC
  lif_async_kernel<<<grid, block, 0, stream>>>(spikes, w, out, T, B);
#else
  lif_plain_kernel<<<grid, block, 0, stream>>>(spikes, w, out, T, B);
#endif
}